// GraphNormWeighted_25202868093051
// MI455X (gfx1250) — compile-verified
//
#include <hip/hip_runtime.h>

typedef float v2f __attribute__((ext_vector_type(2)));
typedef float v4f __attribute__((ext_vector_type(4)));
typedef float v8f __attribute__((ext_vector_type(8)));

#define NGRAPHS 1000
#define CCH     128
#define CHUNK   512
#define EPSF    1e-5f

// ---------------------------------------------------------------- zero scratch
__global__ __launch_bounds__(256) void gn_zero(float* __restrict__ p, int n) {
  int i = blockIdx.x * 256 + threadIdx.x;
  if (i < n) p[i] = 0.0f;
}

// ---------------------------------------------------------------- pass 1: stats
// Block = 256 threads = 8 waves; wave w owns channels [16w, 16w+16).
// Per chunk: find each same-graph run via uniform binary search (batch is
// sorted), then stream the run in 4-node groups with NO segment logic in the
// hot loop: A = node weights (16x4, rows identical), B = x tile (4x16),
// D += A*B accumulates sum(w*x) / sum(w*x^2) for 16 channels in f32.
__global__ __launch_bounds__(256) void gn_stats(
    const float* __restrict__ x, const long long* __restrict__ batch,
    const float* __restrict__ nw,
    float* __restrict__ sums, float* __restrict__ sums2, float* __restrict__ wsum,
    int N)
{
  const int lane = threadIdx.x & 31;
  const int half = lane >> 4;                // 0: lanes 0-15, 1: lanes 16-31
  const int li   = lane & 15;
  const int cb   = (threadIdx.x >> 5) << 4;  // wave * 16 channels
  const int c    = cb + li;
  const int ja   = half ? 2 : 0;             // node slot for b.x
  const int jb   = half ? 3 : 1;             // node slot for b.y

  const int n0base = blockIdx.x * CHUNK;
  if (n0base >= N) return;                   // uniform per block
  const int nend = (n0base + CHUNK < N) ? (n0base + CHUNK) : N;

  int p = n0base;
  while (p < nend) {                         // uniform loop over runs
    const long long gl = batch[p];
    const int gcur = (int)gl;

    // ---- binary search: first index in (p, nend) with batch != gcur
    int lo = p + 1, hi = nend;
    while (lo < hi) {
      int mid = (lo + hi) >> 1;
      if (batch[mid] == gl) lo = mid + 1; else hi = mid;
    }
    const int e = lo;                        // run = [p, e)

    v8f accA = {0.f,0.f,0.f,0.f,0.f,0.f,0.f,0.f};
    v8f accB = accA;
    float wrun = 0.f;

    // strength-reduced stream pointer: xa = xpa[0], xb = xpa[+CCH],
    // advance by 4 nodes (2048 B) per iteration.
    const float* xpa = x + (size_t)(p + ja) * CCH + c;

    // ---- hot loop: full 4-node groups, zero segment logic
    int n = p;
    for (; n + 4 <= e; n += 4, xpa += 4 * CCH) {
      const float wv0 = nw[n],     wv1 = nw[n + 1];
      const float wv2 = nw[n + 2], wv3 = nw[n + 3];
      const float xa = xpa[0];
      const float xb = xpa[CCH];
      __builtin_prefetch(xpa + 8 * CCH, 0, 3);   // speculative, imm offset

      v2f a;  a.x = half ? wv2 : wv0;  a.y = half ? wv3 : wv1;
      v2f bA; bA.x = xa;       bA.y = xb;
      v2f bB; bB.x = xa * xa;  bB.y = xb * xb;

      accA = __builtin_amdgcn_wmma_f32_16x16x4_f32(
          false, a, false, bA, (short)0, accA, false, false);
      accB = __builtin_amdgcn_wmma_f32_16x16x4_f32(
          false, a, false, bB, (short)0, accB, false, false);
      wrun += wv0 + wv1 + wv2 + wv3;
    }

    // ---- masked tail group (0-3 nodes), once per run
    if (n < e) {                             // uniform
      const float wv0 = nw[n];
      const float wv1 = (n + 1 < e) ? nw[n + 1] : 0.f;
      const float wv2 = (n + 2 < e) ? nw[n + 2] : 0.f;
      const float wv3 = (n + 3 < e) ? nw[n + 3] : 0.f;
      const float xa = (n + ja < e) ? xpa[0]   : 0.f;
      const float xb = (n + jb < e) ? xpa[CCH] : 0.f;

      v2f a;  a.x = half ? wv2 : wv0;  a.y = half ? wv3 : wv1;
      v2f bA; bA.x = xa;       bA.y = xb;
      v2f bB; bB.x = xa * xa;  bB.y = xb * xb;

      accA = __builtin_amdgcn_wmma_f32_16x16x4_f32(
          false, a, false, bA, (short)0, accA, false, false);
      accB = __builtin_amdgcn_wmma_f32_16x16x4_f32(
          false, a, false, bB, (short)0, accB, false, false);
      wrun += wv0 + wv1 + wv2 + wv3;
    }

    // ---- flush once per run. All D rows are identical (A rows identical):
    // lanes 0-15 hold row M=0 in acc[0], lanes 16-31 hold row M=8 (same).
    if (half == 0) atomicAdd(&sums [gcur * CCH + c], accA[0]);
    else           atomicAdd(&sums2[gcur * CCH + c], accB[0]);
    if (threadIdx.x == 0) atomicAdd(&wsum[gcur], wrun);

    p = e;
  }
}

// ---------------------------------------------------------------- pass 2: finalize
// sums  <- sub = mean * mean_scale
// sums2 <- inv = weight / sqrt(var + eps),  var = E[x^2] + ms*m^2*(ms-2)
__global__ __launch_bounds__(256) void gn_finalize(
    float* __restrict__ sums, float* __restrict__ sums2,
    const float* __restrict__ wsum, const float* __restrict__ weight,
    const float* __restrict__ mean_scale)
{
  int i = blockIdx.x * 256 + threadIdx.x;
  if (i >= NGRAPHS * CCH) return;
  int b  = i >> 7;
  int ch = i & (CCH - 1);
  float wsc  = fmaxf(wsum[b], 1e-12f);
  float mean = sums[i]  / wsc;
  float e2   = sums2[i] / wsc;
  float ms   = mean_scale[ch];
  float var  = e2 + ms * mean * mean * (ms - 2.0f);
  var = fmaxf(var, 0.0f);
  sums[i]  = mean * ms;                       // sub
  sums2[i] = weight[ch] / sqrtf(var + EPSF);  // inv
}

// ---------------------------------------------------------------- pass 3: apply
// One float4 (4 channels) per thread, coalesced b128; x/out are stream-once
// -> non-temporal; sub/inv (1 MB) stay hot in L2.
__global__ __launch_bounds__(256) void gn_apply(
    const float* __restrict__ x, const long long* __restrict__ batch,
    const float* __restrict__ sub, const float* __restrict__ inv,
    const float* __restrict__ bias, float* __restrict__ out, int N)
{
  size_t idx   = (size_t)blockIdx.x * 256 + threadIdx.x;
  size_t total = (size_t)N * (CCH / 4);
  if (idx >= total) return;
  size_t n  = idx >> 5;                 // CCH/4 == 32 float4 per node
  int    c4 = (int)(idx & 31) << 2;
  int    b  = (int)batch[n];

  v4f xv = __builtin_nontemporal_load((const v4f*)x + idx);
  v4f sv = *(const v4f*)(sub  + (size_t)b * CCH + c4);
  v4f iv = *(const v4f*)(inv  + (size_t)b * CCH + c4);
  v4f bv = *(const v4f*)(bias + c4);
  v4f o  = (xv - sv) * iv + bv;
  __builtin_nontemporal_store(o, (v4f*)out + idx);
}

// ---------------------------------------------------------------- launcher
extern "C" void kernel_launch(void* const* d_in, const int* in_sizes, int n_in,
                              void* d_out, int out_size, void* d_ws, size_t ws_size,
                              hipStream_t stream)
{
  const float*     x      = (const float*)d_in[0];
  const long long* batch  = (const long long*)d_in[1];   // int64
  const float*     nw     = (const float*)d_in[2];
  const float*     weight = (const float*)d_in[3];
  const float*     bias   = (const float*)d_in[4];
  const float*     mscale = (const float*)d_in[5];
  float*           out    = (float*)d_out;
  const int N = in_sizes[1];                             // node count

  // scratch layout: sums[B*C] | sums2[B*C] | wsum[B]   (~1.03 MB)
  float* sums  = (float*)d_ws;
  float* sums2 = sums  + NGRAPHS * CCH;
  float* wsum  = sums2 + NGRAPHS * CCH;
  const int zn = NGRAPHS * CCH * 2 + NGRAPHS;

  gn_zero<<<(zn + 255) / 256, 256, 0, stream>>>(sums, zn);
  gn_stats<<<(N + CHUNK - 1) / CHUNK, 256, 0, stream>>>(
      x, batch, nw, sums, sums2, wsum, N);
  gn_finalize<<<(NGRAPHS * CCH + 255) / 256, 256, 0, stream>>>(
      sums, sums2, wsum, weight, mscale);
  size_t total = (size_t)N * (CCH / 4);
  gn_apply<<<(unsigned)((total + 255) / 256), 256, 0, stream>>>(
      x, batch, sums, sums2, bias, out, N);
}